// FusedThorExpert_57518202028600
// MI455X (gfx1250) — compile-verified
//
#include <hip/hip_runtime.h>

typedef __attribute__((ext_vector_type(2))) float v2f;
typedef __attribute__((ext_vector_type(4))) float v4f;
typedef __attribute__((ext_vector_type(8))) float v8f;

#define HIDDEN 1024
#define INTER  4096
#define NEXP   8
#define TOTAL_TOKENS 12288

#define BM 128
#define BN 128
#define BK 32
#define LPITCH 36   // BK + 4: keeps float4 LDS stores 16B aligned, kills frag-read bank conflicts

__device__ __constant__ int c_tok_offs[NEXP + 1] =
    {0, 512, 1536, 3584, 7680, 8704, 9216, 11264, 12288};

// ---------------------------------------------------------------------------
// Block GEMM core: C[m_base:+128, n_base:+128] = A @ B (+ bias), fp32 WMMA.
//   A: [M, K] row-major, lda
//   BT=true : B element (k,n) = Bglob[n*ldb + k]   (NT GEMM, weights [N,K])
//   BT=false: B element (k,n) = Bglob[k*ldb + n]   (NN GEMM, weights [K,N])
// Requires: Kdim % 32 == 0, tiles fully in-bounds (all dims multiples of 128).
// ---------------------------------------------------------------------------
template <bool BT, bool HAS_BIAS>
__device__ __forceinline__ void gemm_tile(
    const float* __restrict__ A, int lda,
    const float* __restrict__ B, int ldb,
    const float* __restrict__ bias,
    float* __restrict__ C, int ldc,
    int Kdim, int m_base, int n_base)
{
    __shared__ float As[BM * LPITCH];
    __shared__ float Bs[BN * LPITCH];

    const int tid  = threadIdx.x;
    const int lane = tid & 31;
    const int wave = tid >> 5;
    const int wm   = (wave & 1) * 64;   // wave M offset in block (2 waves in M)
    const int wn   = (wave >> 1) * 32;  // wave N offset in block (4 waves in N)
    const int lrow = lane & 15;         // row within 16x16 fragment
    const int lkhi = (lane >> 4) * 2;   // k sub-offset (lanes 16-31 hold K=2,3)

    v8f acc[4][2];
#pragma unroll
    for (int ti = 0; ti < 4; ++ti)
#pragma unroll
        for (int tj = 0; tj < 2; ++tj)
            acc[ti][tj] = (v8f){0.f, 0.f, 0.f, 0.f, 0.f, 0.f, 0.f, 0.f};

    const int lcol4 = (tid & 7) * 4;    // 8 float4 per 32-wide K row
    const int lrow0 = tid >> 3;         // 32 rows per pass, 4 passes -> 128 rows

    for (int kb = 0; kb < Kdim; kb += BK) {
        // ---- stage A tile: As[row][k] = A[m_base+row][kb+k] -------------
#pragma unroll
        for (int p = 0; p < 4; ++p) {
            const int r = lrow0 + p * 32;
            v4f v = *(const v4f*)(A + (size_t)(m_base + r) * lda + kb + lcol4);
            *(v4f*)(&As[r * LPITCH + lcol4]) = v;
        }
        // ---- stage B tile: Bs[n][k] = B(kb+k, n_base+n) -----------------
        if (BT) {
#pragma unroll
            for (int p = 0; p < 4; ++p) {
                const int r = lrow0 + p * 32;
                v4f v = *(const v4f*)(B + (size_t)(n_base + r) * ldb + kb + lcol4);
                *(v4f*)(&Bs[r * LPITCH + lcol4]) = v;
            }
        } else {
            // B row-major [K,N]: read coalesced along n, scatter into LDS
#pragma unroll
            for (int p = 0; p < 16; ++p) {
                const int idx = p * 256 + tid;
                const int kk  = idx >> 7;      // 0..31
                const int nn  = idx & 127;     // 0..127
                Bs[nn * LPITCH + kk] =
                    B[(size_t)(kb + kk) * ldb + (n_base + nn)];
            }
        }
        __syncthreads();

        // ---- math: 8 k-steps of 4, 8 WMMAs per step ---------------------
#pragma unroll
        for (int ks = 0; ks < BK; ks += 4) {
            v2f af[4], bf[2];
#pragma unroll
            for (int ti = 0; ti < 4; ++ti)
                af[ti] = *(const v2f*)(&As[(wm + ti * 16 + lrow) * LPITCH + ks + lkhi]);
#pragma unroll
            for (int tj = 0; tj < 2; ++tj)
                bf[tj] = *(const v2f*)(&Bs[(wn + tj * 16 + lrow) * LPITCH + ks + lkhi]);
#pragma unroll
            for (int ti = 0; ti < 4; ++ti)
#pragma unroll
                for (int tj = 0; tj < 2; ++tj)
                    acc[ti][tj] = __builtin_amdgcn_wmma_f32_16x16x4_f32(
                        /*neg_a=*/false, af[ti], /*neg_b=*/false, bf[tj],
                        /*c_mod=*/(short)0, acc[ti][tj],
                        /*reuse_a=*/false, /*reuse_b=*/false);
        }
        __syncthreads();
    }

    // ---- epilogue: C/D layout: vgpr r, lane l -> (m = r + 8*(l>>4), n = l&15)
    const int mh = (lane >> 4) * 8;
#pragma unroll
    for (int ti = 0; ti < 4; ++ti) {
#pragma unroll
        for (int tj = 0; tj < 2; ++tj) {
            const int n = n_base + wn + tj * 16 + lrow;
            const float bv = HAS_BIAS ? bias[n] : 0.0f;
#pragma unroll
            for (int r = 0; r < 8; ++r) {
                const int m = m_base + wm + ti * 16 + mh + r;
                C[(size_t)m * ldc + n] = acc[ti][tj][r] + bv;
            }
        }
    }
}

// ---------------------------------------------------------------------------
// Kernel 1: W_e = w2_e @ w1_e   ([1024,4096] @ [4096,1024] -> [1024,1024])
// ---------------------------------------------------------------------------
__global__ __launch_bounds__(256) void precompute_W_kernel(
    const float* __restrict__ w1, const float* __restrict__ w2,
    float* __restrict__ Wws)
{
    const int e = blockIdx.z;
    const float* A = w2 + (size_t)e * HIDDEN * INTER;  // [1024, 4096] row-major
    const float* B = w1 + (size_t)e * INTER * HIDDEN;  // [4096, 1024] row-major (NN)
    float* Cc = Wws + (size_t)e * HIDDEN * HIDDEN;
    gemm_tile<false, false>(A, INTER, B, HIDDEN, nullptr, Cc, HIDDEN,
                            INTER, blockIdx.x * BM, blockIdx.y * BN);
}

// ---------------------------------------------------------------------------
// Kernel 2: beff_e = w2_e @ b1_e + b2_e
// ---------------------------------------------------------------------------
__global__ __launch_bounds__(256) void beff_kernel(
    const float* __restrict__ w2, const float* __restrict__ b1,
    const float* __restrict__ b2, float* __restrict__ beff)
{
    const int e = blockIdx.x;
    const float* w2e = w2 + (size_t)e * HIDDEN * INTER;
    const v4f* b1e = (const v4f*)(b1 + (size_t)e * INTER);
    for (int j = threadIdx.x; j < HIDDEN; j += blockDim.x) {
        const v4f* row = (const v4f*)(w2e + (size_t)j * INTER);
        v4f a4 = (v4f){0.f, 0.f, 0.f, 0.f};
        for (int i = 0; i < INTER / 4; ++i) a4 += row[i] * b1e[i];
        beff[(size_t)e * HIDDEN + j] =
            b2[(size_t)e * HIDDEN + j] + a4[0] + a4[1] + a4[2] + a4[3];
    }
}

// ---------------------------------------------------------------------------
// Kernel 3: out = x @ W_eᵀ + beff_e  (per 128-token block; caps are 128-aligned)
// ---------------------------------------------------------------------------
__global__ __launch_bounds__(256) void moe_out_kernel(
    const float* __restrict__ X, const float* __restrict__ Wws,
    const float* __restrict__ beff, float* __restrict__ out)
{
    const int m_base = blockIdx.x * BM;
    int e = 0;
    while (e < NEXP - 1 && m_base >= c_tok_offs[e + 1]) ++e;
    const float* B = Wws + (size_t)e * HIDDEN * HIDDEN;  // [N=1024, K=1024] (NT)
    const float* bias = beff + (size_t)e * HIDDEN;
    gemm_tile<true, true>(X, HIDDEN, B, HIDDEN, bias, out, HIDDEN,
                          HIDDEN, m_base, blockIdx.y * BN);
}

extern "C" void kernel_launch(void* const* d_in, const int* in_sizes, int n_in,
                              void* d_out, int out_size, void* d_ws, size_t ws_size,
                              hipStream_t stream)
{
    const float* x  = (const float*)d_in[0];
    const float* w1 = (const float*)d_in[1];
    const float* b1 = (const float*)d_in[2];
    const float* w2 = (const float*)d_in[3];
    const float* b2 = (const float*)d_in[4];
    float* out = (float*)d_out;

    float* Wws  = (float*)d_ws;                              // 8*1024*1024 fp32 = 33.55 MB
    float* beff = Wws + (size_t)NEXP * HIDDEN * HIDDEN;      // + 8*1024 fp32

    dim3 g1(HIDDEN / BM, HIDDEN / BN, NEXP);                 // 8 x 8 x 8
    precompute_W_kernel<<<g1, 256, 0, stream>>>(w1, w2, Wws);

    beff_kernel<<<NEXP, 256, 0, stream>>>(w2, b1, b2, beff);

    dim3 g2(TOTAL_TOKENS / BM, HIDDEN / BN);                 // 96 x 8
    moe_out_kernel<<<g2, 256, 0, stream>>>(x, Wws, beff, out);
}